// ChannelLatencySeq2Seq_2920577762005
// MI455X (gfx1250) — compile-verified
//
#include <hip/hip_runtime.h>
#include <math.h>

#define B_   16
#define C_   256
#define T_   1024
#define NCOL 2304                 // C_ * 9
#define SZ_BCT (B_ * C_ * T_)     // 4194304

typedef __attribute__((ext_vector_type(2))) float v2f;
typedef __attribute__((ext_vector_type(8))) float v8f;

// ---------------------------------------------------------------------------
// Kernel 1: zero the recon region (t>=9 entries stay 0) and echo x to output.
// ---------------------------------------------------------------------------
__global__ __launch_bounds__(256) void zero_copy_kernel(
    const float* __restrict__ x, float* __restrict__ out)
{
    size_t i = (size_t)blockIdx.x * blockDim.x + threadIdx.x;   // float4 index
    const float4* x4 = (const float4*)x;
    float4* r4 = (float4*)out;                 // recon region
    float4* c4 = (float4*)(out + SZ_BCT);      // x-echo region
    float4 z; z.x = 0.f; z.y = 0.f; z.z = 0.f; z.w = 0.f;
    r4[i] = z;
    c4[i] = x4[i];
}

// ---------------------------------------------------------------------------
// Kernel 2: fused depthwise-conv (collapsed to one width-9 kernel) + reduce
// + LIF first-crossing latency via wave-parallel affine scan.
// One wave32 per (b,c) row; 8 waves per block; x row staged in LDS.
// ---------------------------------------------------------------------------
__global__ __launch_bounds__(256) void fused_conv_lif_kernel(
    const float* __restrict__ x,
    const float* __restrict__ w3, const float* __restrict__ b3,
    const float* __restrict__ w5, const float* __restrict__ b5,
    const float* __restrict__ w9, const float* __restrict__ b9,
    const float* __restrict__ rw, const float* __restrict__ rb,
    const float* __restrict__ latency_scale,
    float* __restrict__ out_lat, float* __restrict__ out_act)
{
    __shared__ float xs[8][T_];
    const int lane = threadIdx.x & 31;
    const int warp = threadIdx.x >> 5;
    const int row  = blockIdx.x * 8 + warp;        // row = b*C_ + c
    const int c    = row & (C_ - 1);

    const float* xrow = x + (size_t)row * T_;
    for (int i = lane; i < T_; i += 32) xs[warp][i] = xrow[i];
    __syncthreads();

    // Combine reduce_w with the three depthwise kernels into one width-9 tap set.
    float rwl[18];
#pragma unroll
    for (int d = 0; d < 18; ++d) rwl[d] = rw[c * 18 + d];

    float comb[9];
#pragma unroll
    for (int tau = 0; tau < 9; ++tau) {
        float s = 0.f;
#pragma unroll
        for (int o = 0; o < 6; ++o)
            s += rwl[12 + o] * w9[(c * 6 + o) * 9 + tau];
        if (tau >= 2 && tau <= 6) {
#pragma unroll
            for (int o = 0; o < 6; ++o)
                s += rwl[6 + o] * w5[(c * 6 + o) * 5 + (tau - 2)];
        }
        if (tau >= 3 && tau <= 5) {
#pragma unroll
            for (int o = 0; o < 6; ++o)
                s += rwl[o] * w3[(c * 6 + o) * 3 + (tau - 3)];
        }
        comb[tau] = s;
    }
    float effb = rb[c];
#pragma unroll
    for (int o = 0; o < 6; ++o)
        effb += rwl[o] * b3[c * 6 + o] + rwl[6 + o] * b5[c * 6 + o]
              + rwl[12 + o] * b9[c * 6 + o];

    const float ALPHA = 0.81873075307798182f;      // exp(-1/5)
    const float ONEMA = 1.f - ALPHA;
    const float A32   = 0.0016615572731739337f;    // ALPHA^32

    // Pass 1: local drive + local affine accumulation (V_out = A32*V_in + bb).
    const int t0 = lane * 32;
    float dv[32];
    float V = 0.f;
    for (int i = 0; i < 32; ++i) {
        int t = t0 + i;
        float s = effb;
#pragma unroll
        for (int tau = 0; tau < 9; ++tau) {
            int xi = t - 4 + tau;
            float xv = (xi >= 0 && xi < T_) ? xs[warp][xi] : 0.f;
            s += comb[tau] * xv;
        }
        dv[i] = s;
        V = ALPHA * V + ONEMA * s;
    }

    // Inclusive affine-composition scan across lanes (Hillis-Steele).
    float a  = A32;
    float bb = V;
#pragma unroll
    for (int off = 1; off < 32; off <<= 1) {
        float pa = __shfl_up(a,  off, 32);
        float pb = __shfl_up(bb, off, 32);
        if (lane >= off) {
            bb = a * pb + bb;     // compose: cur after prev
            a  = a * pa;
        }
    }
    float vstart = __shfl_up(bb, 1, 32);
    if (lane == 0) vstart = 0.f;

    // Pass 2: rescan chunk with true V_start; find first threshold crossing.
    float latf = 1e30f;
    V = vstart;
    for (int i = 0; i < 32; ++i) {
        V = ALPHA * V + ONEMA * dv[i];
        if (V >= 0.01f && latf > 1e29f) latf = (float)(t0 + i);
    }
#pragma unroll
    for (int off = 16; off >= 1; off >>= 1)
        latf = fminf(latf, __shfl_xor(latf, off, 32));
    if (latf > 1e29f) latf = (float)T_;

    if (lane == 0) {
        out_lat[row] = latf;
        float scale = fmaxf(latency_scale[0], 0.001f);
        out_act[row] = expf(-latf / scale);
    }
}

// ---------------------------------------------------------------------------
// Kernel 3: build W[c, j*9+t] = og[j,c] * sum_d fw[j,c,d] * kp[c,d,t]
// (only the nonzero t<9 slab of the padded kernel database).
// ---------------------------------------------------------------------------
__global__ __launch_bounds__(256) void build_w_kernel(
    const float* __restrict__ w3, const float* __restrict__ w5,
    const float* __restrict__ w9, const float* __restrict__ og,
    const float* __restrict__ fw, float* __restrict__ W)
{
    int gid = blockIdx.x * blockDim.x + threadIdx.x;   // < C_ * NCOL
    int c = gid / NCOL;
    int n = gid - c * NCOL;
    int j = n / 9;
    int t = n - j * 9;

    const float* f = fw + ((size_t)j * C_ + c) * 18;
    float s = 0.f;
#pragma unroll
    for (int o = 0; o < 6; ++o) s += f[12 + o] * w9[(c * 6 + o) * 9 + t];
    if (t < 5) {
#pragma unroll
        for (int o = 0; o < 6; ++o) s += f[6 + o] * w5[(c * 6 + o) * 5 + t];
    }
    if (t < 3) {
#pragma unroll
        for (int o = 0; o < 6; ++o) s += f[o] * w3[(c * 6 + o) * 3 + t];
    }
    W[gid] = og[j * C_ + c] * s;
}

// ---------------------------------------------------------------------------
// Kernel 4: recon = act(16x256) @ W(256x2304) via V_WMMA_F32_16X16X4_F32.
// One wave per 16-column N tile; 64 K-steps. Exact fp32 matrix math.
// ---------------------------------------------------------------------------
__global__ __launch_bounds__(32) void recon_gemm_wmma_kernel(
    const float* __restrict__ act, const float* __restrict__ W,
    float* __restrict__ recon)
{
    const int tile = blockIdx.x;          // 0..143
    const int lane = threadIdx.x;         // 0..31
    const int nl   = lane & 15;
    const int hi   = lane >> 4;           // 0: K pair {k0,k0+1}; 1: {k0+2,k0+3}
    const int ncol = tile * 16 + nl;

    v8f acc = {};
    for (int k0 = 0; k0 < C_; k0 += 4) {
        const int ka = k0 + hi * 2;
        v2f a, b;
        // A: 16x4 fp32 fragment (act rows = batch, K = channel)
        a.x = act[nl * C_ + ka];
        a.y = act[nl * C_ + ka + 1];
        // B: 4x16 fp32 fragment (W rows = channel, cols = j*9+t)
        b.x = W[(size_t)ka * NCOL + ncol];
        b.y = W[(size_t)(ka + 1) * NCOL + ncol];
        acc = __builtin_amdgcn_wmma_f32_16x16x4_f32(
                  false, a, false, b, (short)0, acc, false, false);
    }

    // Scatter D (16x16 f32, 8 VGPRs) into recon[b, j, t] with n = j*9 + t.
    const int j = ncol / 9;
    const int t = ncol - j * 9;
    const size_t base = (size_t)j * T_ + t;
#pragma unroll
    for (int r = 0; r < 8; ++r) {
        int m = r + hi * 8;               // batch row
        recon[(size_t)m * (C_ * T_) + base] = acc[r];
    }
}

// ---------------------------------------------------------------------------
extern "C" void kernel_launch(void* const* d_in, const int* in_sizes, int n_in,
                              void* d_out, int out_size, void* d_ws, size_t ws_size,
                              hipStream_t stream) {
    const float* x  = (const float*)d_in[0];
    const float* w3 = (const float*)d_in[1];
    const float* b3 = (const float*)d_in[2];
    const float* w5 = (const float*)d_in[3];
    const float* b5 = (const float*)d_in[4];
    const float* w9 = (const float*)d_in[5];
    const float* b9 = (const float*)d_in[6];
    const float* rw = (const float*)d_in[7];
    const float* rb = (const float*)d_in[8];
    const float* ls = (const float*)d_in[9];
    const float* og = (const float*)d_in[10];
    const float* fw = (const float*)d_in[11];

    float* out   = (float*)d_out;
    float* recon = out;                               // (B,C,T)
    float* lat   = out + 2 * (size_t)SZ_BCT;          // (B,C)
    float* act   = lat + B_ * C_;                     // (B,C)
    float* W     = (float*)d_ws;                      // (C_, NCOL) = 2.36 MB

    // 1) zero recon + echo x  (float4-vectorized; SZ_BCT/4 lanes)
    zero_copy_kernel<<<SZ_BCT / 4 / 256, 256, 0, stream>>>(x, out);

    // 2) fused conv + LIF latency  (4096 rows, one wave32 each)
    fused_conv_lif_kernel<<<(B_ * C_) / 8, 256, 0, stream>>>(
        x, w3, b3, w5, b5, w9, b9, rw, rb, ls, lat, act);

    // 3) build W matrix in workspace
    build_w_kernel<<<(C_ * NCOL) / 256, 256, 0, stream>>>(w3, w5, w9, og, fw, W);

    // 4) recon GEMM on the matrix unit (fp32 WMMA)
    recon_gemm_wmma_kernel<<<NCOL / 16, 32, 0, stream>>>(act, W, recon);
}